// DTWLoss_32701880992228
// MI455X (gfx1250) — compile-verified
//
#include <hip/hip_runtime.h>
#include <math.h>

// DTW over a 2048x2048 cost matrix, cost computed on the fly from s,t in LDS.
// Anti-diagonal wavefront, single 1024-thread workgroup (32 wave32 waves),
// 3 rotating LDS diagonal buffers, one workgroup barrier per diagonal.
// CDNA5 paths: global_load_async_to_lds_b64 (+ s_wait_asynccnt) for staging;
// split workgroup barriers (s_barrier_signal/-wait) per diagonal.
//
// Register-carried recurrence: the "up" neighbor F[i-1][j] is the value this
// thread itself computed last diagonal for the same column -> kept in a VGPR,
// and sh_t[j] is loop-invariant per thread -> hoisted. Per cell: 3 LDS loads
// (s[i], left, diag) + 1 LDS store, down from 5 loads.

#define M_ROWS 2048
#define N_COLS 2048
#define NT     1024

__global__ __launch_bounds__(NT, 1)
void dtw_wavefront_kernel(const float* __restrict__ s,
                          const float* __restrict__ t,
                          float* __restrict__ out) {
    __shared__ float sh_s[M_ROWS];
    __shared__ float sh_t[N_COLS];
    __shared__ float diag_buf[3][N_COLS];

    const int tid = threadIdx.x;

    // ---- Stage s,t into LDS via CDNA5 async copy engine (ASYNCcnt path). ----
    // Each lane pulls 8B (two floats) of each array: 1024 lanes * 2 = 2048 elems.
    {
        // Low 32 bits of a flat shared-aperture address are the LDS byte offset
        // (shared aperture base occupies addr[63:32] on CDNA5).
        unsigned lds_s = (unsigned)(size_t)(&sh_s[2 * tid]);
        unsigned lds_t = (unsigned)(size_t)(&sh_t[2 * tid]);
        const float* gs = s + 2 * tid;
        const float* gt = t + 2 * tid;
        asm volatile("global_load_async_to_lds_b64 %0, %1, off"
                     :
                     : "v"(lds_s), "v"(gs)
                     : "memory");
        asm volatile("global_load_async_to_lds_b64 %0, %1, off"
                     :
                     : "v"(lds_t), "v"(gt)
                     : "memory");
        // Wait for this wave's async transfers to land in LDS.
        asm volatile("s_wait_asynccnt 0" ::: "memory");
    }
    __syncthreads();  // make LDS from all waves' async copies visible

    // ---- Wavefront over anti-diagonals d = i + j. ----
    // Buffer roles per iteration: cur = diag d, p1 = diag d-1, p2 = diag d-2.
    float* cur = &diag_buf[0][0];
    float* p1  = &diag_buf[1][0];
    float* p2  = &diag_buf[2][0];

    const int j0 = tid;         // columns owned by this thread (stride-NT:
    const int j1 = tid + NT;    //  conflict-free LDS access within a wave)

    // Loop-invariant target values for this thread's two columns.
    const float tj0 = sh_t[j0];
    const float tj1 = sh_t[j1];

    // Register-carried "up" values: F[i-1][j0] and F[i-1][j1].
    // Init 0 == boundary value for the first active diagonal (i == 0).
    float v0 = 0.0f;
    float v1 = 0.0f;

    const int NDIAG = M_ROWS + N_COLS - 1;  // 4095

    for (int d = 0; d < NDIAG; ++d) {
        // --- cell (i, j0) ---
        {
            const int i = d - j0;
            if ((unsigned)i < (unsigned)M_ROWS) {
                // left  = F[i][j0-1]   (neighbor thread, diag d-1)
                // dgv   = F[i-1][j0-1] (neighbor thread, diag d-2)
                // up    = F[i-1][j0]   (this thread, register v0)
                const float left = (j0 > 0) ? p1[j0 - 1] : 0.0f;
                const float dgv  = (i > 0 && j0 > 0) ? p2[j0 - 1] : 0.0f;
                const float c    = fabsf(sh_s[i] - tj0);
                v0 = c + fminf(fminf(v0, left), dgv);
                cur[j0] = v0;
            }
        }
        // --- cell (i, j1) ---  (j1 >= 1024, so j1 > 0 statically)
        {
            const int i = d - j1;
            if ((unsigned)i < (unsigned)M_ROWS) {
                const float left = p1[j1 - 1];
                const float dgv  = (i > 0) ? p2[j1 - 1] : 0.0f;
                const float c    = fabsf(sh_s[i] - tj1);
                v1 = c + fminf(fminf(v1, left), dgv);
                cur[j1] = v1;
                // Final cell (M-1, N-1) lies on the last diagonal at j = N-1
                // (owned by tid == NT-1). Reference returns F[M-1][N-1].
                if (d == NDIAG - 1 && j1 == N_COLS - 1) {
                    *out = v1;
                }
            }
        }

        __syncthreads();

        // Rotate: next iteration sees p1 = this cur, p2 = this p1,
        // and reuses this p2's storage for its cur (last read is 1 barrier
        // before its next write -> safe).
        float* tmp = p2;
        p2 = p1;
        p1 = cur;
        cur = tmp;
    }
}

extern "C" void kernel_launch(void* const* d_in, const int* in_sizes, int n_in,
                              void* d_out, int out_size, void* d_ws, size_t ws_size,
                              hipStream_t stream) {
    (void)in_sizes; (void)n_in; (void)out_size; (void)d_ws; (void)ws_size;
    const float* s = (const float*)d_in[0];   // source, 2048 f32
    const float* t = (const float*)d_in[1];   // target, 2048 f32
    float* out = (float*)d_out;               // scalar f32 result

    dtw_wavefront_kernel<<<1, NT, 0, stream>>>(s, t, out);
}